// RGCNModel_59004260713107
// MI455X (gfx1250) — compile-verified
//
#include <hip/hip_runtime.h>

// ---------------------------------------------------------------------------
// RGCN (2-layer) for MI455X / gfx1250.
//   N=50000 nodes, E=800000 edges, IN=H=128, R=8 relations, OUT=2.
//
// Mean-aggregation commutes with the per-relation linear map, so:
//   out[v] = bias + X[v]@root + sum_r (mean-agg_r X)[v] @ W[r]
// Fused path (if workspace >= ~258 MB):
//   one edge pass scatters into agg8[N, R*H]; a single WMMA GEMM with
//   K = R*H + H keeps C in registers across all chunks (C written once).
// Fallback path: per-relation agg buffer + accumulate-GEMMs (smaller ws).
// Exact fp32 matrix math via V_WMMA_F32_16X16X4_F32.
// ---------------------------------------------------------------------------

#define GN 50000
#define GE 800000
#define GH 128
#define GR 8
#define BN_EPS 1e-5f
#define LDSPAD 132  // padded LDS row stride (floats): conflict-free b64 frags

typedef __attribute__((ext_vector_type(2))) float v2f;
typedef __attribute__((ext_vector_type(8))) float v8f;

__device__ __forceinline__ float fatomic_add(float* p, float v) {
  return unsafeAtomicAdd(p, v);  // hardware global_atomic_add_f32
}

// ---------------- utility kernels -------------------------------------------

__global__ void zero_kernel(float* __restrict__ p, long long n) {
  long long i = (long long)blockIdx.x * blockDim.x + threadIdx.x;
  long long stride = (long long)gridDim.x * blockDim.x;
  for (; i < n; i += stride) p[i] = 0.0f;
}

__global__ void count_deg_kernel(const int* __restrict__ tgt,
                                 const int* __restrict__ et,
                                 float* __restrict__ deg) {
  int e = blockIdx.x * blockDim.x + threadIdx.x;
  if (e < GE) fatomic_add(&deg[tgt[e] * GR + et[e]], 1.0f);
}

__global__ void invert_deg_kernel(float* __restrict__ deg) {
  int i = blockIdx.x * blockDim.x + threadIdx.x;
  if (i < GN * GR) {
    float d = deg[i];
    deg[i] = (d > 0.0f) ? (1.0f / d) : 0.0f;
  }
}

__global__ void init_bias_kernel(float* __restrict__ acc,
                                 const float* __restrict__ b) {
  int i = blockIdx.x * blockDim.x + threadIdx.x;
  if (i < GN * GH) acc[i] = b[i & (GH - 1)];
}

// ---------------- edge scatter ----------------------------------------------

// Fused: one pass, agg8[t, r, c] += inv_deg * X[s, c].  One block per edge.
__global__ void scatter_all_kernel(const float* __restrict__ X,
                                   const int* __restrict__ src,
                                   const int* __restrict__ tgt,
                                   const int* __restrict__ et,
                                   const float* __restrict__ invdeg,
                                   float* __restrict__ agg8) {
  int e = blockIdx.x;
  int r = et[e];
  int s = src[e];
  int t = tgt[e];
  float w = invdeg[t * GR + r];
  int c = threadIdx.x;  // 128 channels
  fatomic_add(&agg8[(size_t)t * (GR * GH) + r * GH + c], X[s * GH + c] * w);
}

// Fallback: relation-filtered pass into agg[N, H].
__global__ void scatter_rel_kernel(const float* __restrict__ X,
                                   const int* __restrict__ src,
                                   const int* __restrict__ tgt,
                                   const int* __restrict__ et,
                                   const float* __restrict__ invdeg,
                                   float* __restrict__ agg, int r) {
  int e = blockIdx.x;
  if (et[e] != r) return;
  int s = src[e];
  int t = tgt[e];
  float w = invdeg[t * GR + r];
  int c = threadIdx.x;
  fatomic_add(&agg[t * GH + c], X[s * GH + c] * w);
}

// ---------------- fused WMMA GEMM -------------------------------------------
// acc[v,:] = bias + X[v,:]@root + sum_r agg8[v,r,:]@W[r]
// 8 waves/block, each wave owns a 16-row stripe x all 8 column tiles.
// Per K-chunk (9 chunks of 128): stage A chunk to LDS (coalesced float4 ->
// ds_store_b128), then 32 WMMA k-steps reading conflict-free b64 fragments.
// Fragment layouts per CDNA5 ISA 7.12.2 (16x16x4 f32).
__global__ __launch_bounds__(256) void rgcn_gemm_kernel(
    const float* __restrict__ Agg,   // [N, R*GH]
    const float* __restrict__ X,     // [N, GH]
    const float* __restrict__ W,     // [R, GH, GH]
    const float* __restrict__ root,  // [GH, GH]
    const float* __restrict__ bias,  // [GH]
    float* __restrict__ C, int M) {
  extern __shared__ float As[];  // 128 rows x LDSPAD floats
  const int lane = threadIdx.x & 31;
  const int wave = threadIdx.x >> 5;
  const int m_blk = blockIdx.x * 128;
  const int m0 = m_blk + wave * 16;
  const bool active = (m0 < M);  // wave-uniform; M % 16 == 0
  const int half = lane >> 4;
  const int l15 = lane & 15;
  const int koff = half * 2;
  const int arow_l = wave * 16 + l15;  // local A row for this lane's fragment
  const int crow = m0 + half * 8;

  v8f acc[8];
#pragma unroll
  for (int nt = 0; nt < 8; ++nt) {
    float b0 = bias[nt * 16 + l15];
#pragma unroll
    for (int j = 0; j < 8; ++j) acc[nt][j] = b0;
  }

  const int srow = threadIdx.x >> 5;        // staging: wave stages 1 row/sweep
  const int scol = (threadIdx.x & 31) * 4;  // 32 lanes x float4 = 128 floats

  for (int kc = 0; kc < 9; ++kc) {
    const float* Asrc;
    int lda;
    const float* Bsrc;
    if (kc < 8) {
      Asrc = Agg + kc * GH;
      lda = GR * GH;
      Bsrc = W + kc * GH * GH;
    } else {
      Asrc = X;
      lda = GH;
      Bsrc = root;
    }
    __syncthreads();  // previous chunk's LDS reads done
#pragma unroll
    for (int it = 0; it < 16; ++it) {
      int row = srow + it * 8;  // 0..127
      int g = m_blk + row;
      float4 v = make_float4(0.0f, 0.0f, 0.0f, 0.0f);
      if (g < M) v = *(const float4*)(Asrc + (size_t)g * lda + scol);
      *(float4*)&As[row * LDSPAD + scol] = v;
    }
    __syncthreads();
    if (active) {
      for (int kb = 0; kb < 32; ++kb) {
        const int k = kb * 4 + koff;
        v2f a = *(const v2f*)&As[arow_l * LDSPAD + k];  // ds_load_b64
        const float* Brow0 = Bsrc + k * GH;
        const float* Brow1 = Bsrc + (k + 1) * GH;
#pragma unroll
        for (int nt = 0; nt < 8; ++nt) {
          v2f b;
          b.x = Brow0[nt * 16 + l15];
          b.y = Brow1[nt * 16 + l15];
          acc[nt] = __builtin_amdgcn_wmma_f32_16x16x4_f32(
              false, a, false, b, (short)0, acc[nt], false, false);
        }
      }
    }
  }

  if (active) {
#pragma unroll
    for (int nt = 0; nt < 8; ++nt)
#pragma unroll
      for (int j = 0; j < 8; ++j)
        C[(size_t)(crow + j) * GH + nt * 16 + l15] = acc[nt][j];
  }
}

// ---------------- fallback WMMA GEMM: C += A @ B ----------------------------
__global__ __launch_bounds__(256) void gemm_acc_kernel(
    const float* __restrict__ A, const float* __restrict__ B,
    float* __restrict__ C, int M) {
  const int lane = threadIdx.x & 31;
  const int wave = threadIdx.x >> 5;
  const int m0 = (blockIdx.x * 8 + wave) * 16;
  if (m0 >= M) return;
  const int half = lane >> 4;
  const int l15 = lane & 15;
  const int arow = m0 + l15;
  const int koff = half * 2;
  const int crow = m0 + half * 8;

  v8f acc[8];
#pragma unroll
  for (int nt = 0; nt < 8; ++nt)
#pragma unroll
    for (int j = 0; j < 8; ++j)
      acc[nt][j] = C[(crow + j) * GH + nt * 16 + l15];

  for (int kb = 0; kb < 32; ++kb) {
    const int k = kb * 4 + koff;
    v2f a;
    a.x = A[arow * GH + k];
    a.y = A[arow * GH + k + 1];
    const float* Brow0 = B + k * GH;
    const float* Brow1 = B + (k + 1) * GH;
#pragma unroll
    for (int nt = 0; nt < 8; ++nt) {
      v2f b;
      b.x = Brow0[nt * 16 + l15];
      b.y = Brow1[nt * 16 + l15];
      acc[nt] = __builtin_amdgcn_wmma_f32_16x16x4_f32(
          false, a, false, b, (short)0, acc[nt], false, false);
    }
  }

#pragma unroll
  for (int nt = 0; nt < 8; ++nt)
#pragma unroll
    for (int j = 0; j < 8; ++j)
      C[(crow + j) * GH + nt * 16 + l15] = acc[nt][j];
}

// ---------------- BatchNorm (training mode, biased var) + ReLU --------------

__global__ void bn_stats_kernel(const float* __restrict__ x,
                                float* __restrict__ stats) {
  int c = threadIdx.x;  // 128 threads
  float s = 0.0f, ss = 0.0f;
  for (int v = blockIdx.x; v < GN; v += gridDim.x) {
    float t = x[v * GH + c];
    s += t;
    ss += t * t;
  }
  fatomic_add(&stats[c], s);
  fatomic_add(&stats[GH + c], ss);
}

__global__ void bn_apply_relu_kernel(const float* __restrict__ x,
                                     const float* __restrict__ gamma,
                                     const float* __restrict__ beta,
                                     const float* __restrict__ stats,
                                     float* __restrict__ y) {
  int i = blockIdx.x * blockDim.x + threadIdx.x;
  int stride = gridDim.x * blockDim.x;
  const float invn = 1.0f / (float)GN;
  for (; i < GN * GH; i += stride) {
    int c = i & (GH - 1);
    float mean = stats[c] * invn;
    float var = stats[GH + c] * invn - mean * mean;
    float t = gamma[c] * (x[i] - mean) * rsqrtf(var + BN_EPS) + beta[c];
    y[i] = t > 0.0f ? t : 0.0f;
  }
}

// ---------------- classifier: out[v,:] = h[v,:] @ cw + cb -------------------

__global__ void classifier_kernel(const float* __restrict__ h,
                                  const float* __restrict__ cw,
                                  const float* __restrict__ cb,
                                  float* __restrict__ out) {
  int v = blockIdx.x * blockDim.x + threadIdx.x;
  if (v >= GN) return;
  float a0 = cb[0], a1 = cb[1];
  const float* hv = h + v * GH;
#pragma unroll 8
  for (int c = 0; c < GH; ++c) {
    float t = hv[c];
    a0 += t * cw[c * 2 + 0];
    a1 += t * cw[c * 2 + 1];
  }
  out[v * 2 + 0] = a0;
  out[v * 2 + 1] = a1;
}

// ---------------------------------------------------------------------------

extern "C" void kernel_launch(void* const* d_in, const int* in_sizes, int n_in,
                              void* d_out, int out_size, void* d_ws,
                              size_t ws_size, hipStream_t stream) {
  const float* x = (const float*)d_in[0];
  const int* eidx = (const int*)d_in[1];  // [2, E]
  const int* etype = (const int*)d_in[2];
  const float* w1 = (const float*)d_in[3];
  const float* root1 = (const float*)d_in[4];
  const float* b1 = (const float*)d_in[5];
  const float* g1 = (const float*)d_in[6];
  const float* be1 = (const float*)d_in[7];
  const float* w2 = (const float*)d_in[8];
  const float* root2 = (const float*)d_in[9];
  const float* b2 = (const float*)d_in[10];
  const float* g2 = (const float*)d_in[11];
  const float* be2 = (const float*)d_in[12];
  const float* cw = (const float*)d_in[13];
  const float* cb = (const float*)d_in[14];
  float* out = (float*)d_out;

  const int* src = eidx;
  const int* tgt = eidx + GE;

  const long long NH = (long long)GN * GH;
  const long long NRH = (long long)GN * GR * GH;
  const int gemm_grid = (GN + 127) / 128;
  const size_t lds_bytes = (size_t)GH * LDSPAD * sizeof(float);

  const size_t need_fused =
      ((size_t)GN * GR + 2 * (size_t)NH + 256 + (size_t)NRH) * sizeof(float);

  float* ws = (float*)d_ws;

  if (ws_size >= need_fused) {
    // ===================== fused path =====================
    float* invdeg = ws;                  // N*R
    float* acc = invdeg + GN * GR;       // N*H
    float* h = acc + NH;                 // N*H
    float* stats = h + NH;               // 2*H
    float* agg8 = stats + 256;           // N*R*H

    zero_kernel<<<(GN * GR + 255) / 256, 256, 0, stream>>>(invdeg, GN * GR);
    count_deg_kernel<<<(GE + 255) / 256, 256, 0, stream>>>(tgt, etype, invdeg);
    invert_deg_kernel<<<(GN * GR + 255) / 256, 256, 0, stream>>>(invdeg);

    // layer 1
    zero_kernel<<<8192, 256, 0, stream>>>(agg8, NRH);
    scatter_all_kernel<<<GE, GH, 0, stream>>>(x, src, tgt, etype, invdeg, agg8);
    rgcn_gemm_kernel<<<gemm_grid, 256, lds_bytes, stream>>>(agg8, x, w1, root1,
                                                            b1, acc, GN);
    zero_kernel<<<1, 256, 0, stream>>>(stats, 2 * GH);
    bn_stats_kernel<<<512, GH, 0, stream>>>(acc, stats);
    bn_apply_relu_kernel<<<4096, 256, 0, stream>>>(acc, g1, be1, stats, h);

    // layer 2
    zero_kernel<<<8192, 256, 0, stream>>>(agg8, NRH);
    scatter_all_kernel<<<GE, GH, 0, stream>>>(h, src, tgt, etype, invdeg, agg8);
    rgcn_gemm_kernel<<<gemm_grid, 256, lds_bytes, stream>>>(agg8, h, w2, root2,
                                                            b2, acc, GN);
    zero_kernel<<<1, 256, 0, stream>>>(stats, 2 * GH);
    bn_stats_kernel<<<512, GH, 0, stream>>>(acc, stats);
    bn_apply_relu_kernel<<<4096, 256, 0, stream>>>(acc, g2, be2, stats, h);

    classifier_kernel<<<(GN + 127) / 128, 128, 0, stream>>>(h, cw, cb, out);
  } else {
    // ===================== fallback path (small workspace) =====================
    float* invdeg = ws;                  // N*R
    float* agg = invdeg + GN * GR;       // N*H (also reused as h2)
    float* acc = agg + NH;               // N*H
    float* h = acc + NH;                 // N*H
    float* stats = h + NH;               // 2*H

    zero_kernel<<<(GN * GR + 255) / 256, 256, 0, stream>>>(invdeg, GN * GR);
    count_deg_kernel<<<(GE + 255) / 256, 256, 0, stream>>>(tgt, etype, invdeg);
    invert_deg_kernel<<<(GN * GR + 255) / 256, 256, 0, stream>>>(invdeg);

    // layer 1
    init_bias_kernel<<<(int)((NH + 255) / 256), 256, 0, stream>>>(acc, b1);
    gemm_acc_kernel<<<gemm_grid, 256, 0, stream>>>(x, root1, acc, GN);
    for (int r = 0; r < GR; ++r) {
      zero_kernel<<<4096, 256, 0, stream>>>(agg, NH);
      scatter_rel_kernel<<<GE, GH, 0, stream>>>(x, src, tgt, etype, invdeg,
                                                agg, r);
      gemm_acc_kernel<<<gemm_grid, 256, 0, stream>>>(agg, w1 + r * GH * GH,
                                                     acc, GN);
    }
    zero_kernel<<<1, 256, 0, stream>>>(stats, 2 * GH);
    bn_stats_kernel<<<512, GH, 0, stream>>>(acc, stats);
    bn_apply_relu_kernel<<<4096, 256, 0, stream>>>(acc, g1, be1, stats, h);

    // layer 2
    init_bias_kernel<<<(int)((NH + 255) / 256), 256, 0, stream>>>(acc, b2);
    gemm_acc_kernel<<<gemm_grid, 256, 0, stream>>>(h, root2, acc, GN);
    for (int r = 0; r < GR; ++r) {
      zero_kernel<<<4096, 256, 0, stream>>>(agg, NH);
      scatter_rel_kernel<<<GE, GH, 0, stream>>>(h, src, tgt, etype, invdeg,
                                                agg, r);
      gemm_acc_kernel<<<gemm_grid, 256, 0, stream>>>(agg, w2 + r * GH * GH,
                                                     acc, GN);
    }
    zero_kernel<<<1, 256, 0, stream>>>(stats, 2 * GH);
    bn_stats_kernel<<<512, GH, 0, stream>>>(acc, stats);
    bn_apply_relu_kernel<<<4096, 256, 0, stream>>>(acc, g2, be2, stats, agg);

    classifier_kernel<<<(GN + 127) / 128, 128, 0, stream>>>(agg, cw, cb, out);
  }
}